// D3PM_3788161155361
// MI455X (gfx1250) — compile-verified
//
#include <hip/hip_runtime.h>
#include <stdint.h>

// ---------------- problem constants (match reference) ----------------
#define TDIF   500        // diffusion steps; alpha table has TDIF+1 entries
#define NB     256        // batch
#define LSEQ   1024       // sequence length
#define NSV    516        // structure vocab
#define NQV    33         // sequence vocab
#define MASK_S 3
#define MASK_Q 32
#define EPSF   1e-10f

// ---------------- JAX threefry2x32 (bit-exact) ----------------
__host__ __device__ inline void threefry2x32(uint32_t k0, uint32_t k1,
                                             uint32_t x0, uint32_t x1,
                                             uint32_t* o0, uint32_t* o1) {
  uint32_t ks2 = k0 ^ k1 ^ 0x1BD11BDAu;
#define TF_ROTL(v, s) (((v) << (s)) | ((v) >> (32 - (s))))
#define TF_R4(a, b, c, d)                                        \
  x0 += x1; x1 = TF_ROTL(x1, a); x1 ^= x0;                       \
  x0 += x1; x1 = TF_ROTL(x1, b); x1 ^= x0;                       \
  x0 += x1; x1 = TF_ROTL(x1, c); x1 ^= x0;                       \
  x0 += x1; x1 = TF_ROTL(x1, d); x1 ^= x0;
  x0 += k0;  x1 += k1;
  TF_R4(13, 15, 26, 6)  x0 += k1;  x1 += ks2 + 1u;
  TF_R4(17, 29, 16, 24) x0 += ks2; x1 += k0 + 2u;
  TF_R4(13, 15, 26, 6)  x0 += k0;  x1 += k1 + 3u;
  TF_R4(17, 29, 16, 24) x0 += k1;  x1 += ks2 + 4u;
  TF_R4(13, 15, 26, 6)  x0 += ks2; x1 += k0 + 5u;
  *o0 = x0; *o1 = x1;
#undef TF_R4
#undef TF_ROTL
}

// JAX uniform(minval=tiny, maxval=1) from the top 23 random bits, then gumbel.
__device__ __forceinline__ float jax_u01(uint32_t mant /* = bits >> 9 */) {
  const float TINY = 1.1754943508222875e-38f;  // finfo(float32).tiny
  float u = __uint_as_float(mant | 0x3f800000u) - 1.0f;
  u = u * (1.0f - TINY) + TINY;
  return fmaxf(TINY, u);
}
__device__ __forceinline__ float jax_gumbel(uint32_t mant) {
  return -logf(-logf(jax_u01(mant)));
}

// argmax over {best-other, x0, mask} with JAX first-index tie-breaking.
__device__ __forceinline__ float pick_category(int x0, int maskIdx, float a,
                                               uint32_t bestU, int bestN,
                                               uint32_t ux, uint32_t um) {
  const float logEps = logf(EPSF);
  int   wi = bestN;
  float ws = logEps + jax_gumbel(bestU);
  if (x0 == maskIdx) {
    float sM = logf(1.0f + EPSF) + jax_gumbel(ux);   // alpha + (1-alpha) = 1
    if (sM > ws || (sM == ws && maskIdx < wi)) { ws = sM; wi = maskIdx; }
  } else {
    float sX = logf(a + EPSF) + jax_gumbel(ux);
    float sM = logf((1.0f - a) + EPSF) + jax_gumbel(um);
    if (sX > ws || (sX == ws && x0 < wi))      { ws = sX; wi = x0; }
    if (sM > ws || (sM == ws && maskIdx < wi)) { ws = sM; wi = maskIdx; }
  }
  return (float)wi;
}

// -------- CDNA5 async global->LDS staging (gfx1250 path) --------
#if defined(__HIP_DEVICE_COMPILE__) && \
    __has_builtin(__builtin_amdgcn_global_load_async_to_lds_b32)
#define USE_ASYNC_LDS 1
#else
#define USE_ASYNC_LDS 0
#endif

__global__ __launch_bounds__(256)
void d3pm_qsample_kernel(const int* __restrict__ tok, const int* __restrict__ t,
                         const float* __restrict__ alpha, float* __restrict__ out,
                         uint32_t k0, uint32_t k1, int N, int maskIdx, int halfTok) {
  __shared__ float s_alpha[TDIF + 12];  // 512 floats

  const int tid = threadIdx.x;
#if USE_ASYNC_LDS
  for (int j = tid; j <= TDIF; j += 256) {
    __builtin_amdgcn_global_load_async_to_lds_b32(
        (int*)(alpha + j), (int*)&s_alpha[j], 0, 0);
  }
#if __has_builtin(__builtin_amdgcn_s_wait_asynccnt)
  __builtin_amdgcn_s_wait_asynccnt(0);
#else
  asm volatile("s_wait_asynccnt 0" ::: "memory");
#endif
#else
  for (int j = tid; j <= TDIF; j += 256) s_alpha[j] = alpha[j];
#endif
  __syncthreads();

  const int i = blockIdx.x * 256 + tid;   // token pair (i, i + halfTok)
  if (i >= halfTok) return;
  const int iB = i + halfTok;

  __builtin_prefetch(tok + i, 0, 1);
  __builtin_prefetch(tok + iB, 0, 1);

  const int xA = tok[i];
  const int xB = tok[iB];

  // JAX threefry counter layout: flat j pairs with j + M/2; M/2 = halfTok*N,
  // so counter pair (i*N + n, i*N + n + halfTok*N) yields category n of both
  // tokens in one threefry call.
  const uint32_t base  = (uint32_t)i * (uint32_t)N;
  const uint32_t halfM = (uint32_t)halfTok * (uint32_t)N;

  uint32_t bestA = 0u, bestB = 0u, uxA = 0u, umA = 0u, uxB = 0u, umB = 0u;
  int bnA = -1, bnB = -1;

  for (int n = 0; n < N; ++n) {
    uint32_t ra, rb;
    threefry2x32(k0, k1, base + (uint32_t)n, base + (uint32_t)n + halfM, &ra, &rb);
    const uint32_t ua = ra >> 9, ub = rb >> 9;   // 23-bit mantissae; u monotone in these
    if (n == xA)            uxA = ua;
    else if (n == maskIdx)  umA = ua;
    else if (bnA < 0 || ua > bestA) { bestA = ua; bnA = n; }
    if (n == xB)            uxB = ub;
    else if (n == maskIdx)  umB = ub;
    else if (bnB < 0 || ub > bestB) { bestB = ub; bnB = n; }
  }

  const float aA = s_alpha[t[i  / LSEQ]];
  const float aB = s_alpha[t[iB / LSEQ]];
  out[i]  = pick_category(xA, maskIdx, aA, bestA, bnA, uxA, umA);
  out[iB] = pick_category(xB, maskIdx, aB, bestB, bnB, uxB, umB);
}

__global__ void d3pm_copy_t_kernel(const int* __restrict__ t, float* __restrict__ out) {
  const int i = blockIdx.x * blockDim.x + threadIdx.x;
  if (i < NB) out[i] = (float)t[i];
}

extern "C" void kernel_launch(void* const* d_in, const int* in_sizes, int n_in,
                              void* d_out, int out_size, void* d_ws, size_t ws_size,
                              hipStream_t stream) {
  (void)in_sizes; (void)n_in; (void)out_size; (void)d_ws; (void)ws_size;
  const int*   structure = (const int*)d_in[0];
  const int*   sequence  = (const int*)d_in[1];
  const int*   t         = (const int*)d_in[2];
  const float* alpha     = (const float*)d_in[3];
  float*       out       = (float*)d_out;

  // Replicate: key = jax.random.key(42) -> [0,42]; ks, kq = split(key).
  // split bits = threefry([0,42], [0,1,2,3]) -> ks=(f(0,2).x, f(1,3).x),
  //                                             kq=(f(0,2).y, f(1,3).y)
  uint32_t a0, a1, b0, b1;
  threefry2x32(0u, 42u, 0u, 2u, &a0, &a1);
  threefry2x32(0u, 42u, 1u, 3u, &b0, &b1);

  const int halfTok = NB * LSEQ / 2;          // 131072 token pairs
  dim3 blk(256), grd(halfTok / 256);
  d3pm_qsample_kernel<<<grd, blk, 0, stream>>>(structure, t, alpha, out,
                                               a0, b0, NSV, MASK_S, halfTok);
  d3pm_qsample_kernel<<<grd, blk, 0, stream>>>(sequence, t, alpha, out + NB * LSEQ,
                                               a1, b1, NQV, MASK_Q, halfTok);
  d3pm_copy_t_kernel<<<1, 256, 0, stream>>>(t, out + 2 * NB * LSEQ);
}